// GeneratorBlock_8658654069299
// MI455X (gfx1250) — compile-verified
//
#include <hip/hip_runtime.h>

typedef __attribute__((ext_vector_type(2))) float v2f;
typedef __attribute__((ext_vector_type(8))) float v8f;

constexpr int NB = 4, H = 512, W = 512;
constexpr float EPSV = 1e-8f, SLOPE = 0.01f;

// ---------------------------------------------------------------------------
// Prep: styles s = y@A^T + b, demod d = rsqrt(sum (s*w)^2 + eps), and the
// per-sample folded weights w' = d*s*w written in a WMMA-A-fragment-friendly
// layout: [n][kk=kh*3+kw][ci/2][COUT][2]  (ci pair interleaved -> one b64/lane)
// ---------------------------------------------------------------------------
__global__ __launch_bounds__(256) void prep_kernel(
    const float* __restrict__ y,
    const float* __restrict__ a1w, const float* __restrict__ a1b,
    const float* __restrict__ c1w,
    const float* __restrict__ a2w, const float* __restrict__ a2b,
    const float* __restrict__ c2w,
    float* __restrict__ w1p, float* __restrict__ w2p)
{
  __shared__ float s1[64], d1[64], s2[32], d2[32];
  const int n = blockIdx.x, t = threadIdx.x;
  const float* yn = y + n * 512;

  if (t < 64) {
    const float* wr = a1w + t * 512;
    float acc = 0.f;
    for (int k = 0; k < 512; ++k) acc += yn[k] * wr[k];
    s1[t] = acc + a1b[t];
  } else if (t < 96) {
    const int j = t - 64;
    const float* wr = a2w + j * 512;
    float acc = 0.f;
    for (int k = 0; k < 512; ++k) acc += yn[k] * wr[k];
    s2[j] = acc + a2b[j];
  }
  __syncthreads();

  if (t < 64) {
    // conv1: input-channel modulation (shapes consistent in reference)
    float acc = 0.f;
    const float* wo = c1w + t * (64 * 9);
    for (int i = 0; i < 64; ++i) {
      float ws = 0.f;
      for (int k = 0; k < 9; ++k) { const float v = wo[i * 9 + k]; ws += v * v; }
      acc += s1[i] * s1[i] * ws;
    }
    d1[t] = rsqrtf(acc + EPSV);
  } else if (t < 96) {
    // conv2: s2 is [N,32]; reference einsum is shape-inconsistent with 64 input
    // channels, so fold s2 on the (matching) output-channel axis.
    const int j = t - 64;
    float ws = 0.f;
    const float* wo = c2w + j * (64 * 9);
    for (int k = 0; k < 576; ++k) { const float v = wo[k]; ws += v * v; }
    d2[j] = rsqrtf(s2[j] * s2[j] * ws + EPSV);
  }
  __syncthreads();

  // w1' = d1[o]*s1[ci]*w1[o,ci,kh,kw]
  for (int idx = t; idx < 9 * 64 * 64; idx += 256) {
    const int kk = idx >> 12, rem = idx & 4095;
    const int ci = rem >> 6, o = rem & 63;
    const float v = d1[o] * s1[ci] * c1w[o * 576 + ci * 9 + kk];
    w1p[(size_t)n * 36864 + (((kk * 32 + (ci >> 1)) * 64 + o) << 1) + (ci & 1)] = v;
  }
  // w2' = d2[o]*s2[o]*w2[o,ci,kh,kw]
  for (int idx = t; idx < 9 * 64 * 32; idx += 256) {
    const int kk = idx >> 11, rem = idx & 2047;
    const int ci = rem >> 5, o = rem & 31;
    const float v = d2[o] * s2[o] * c2w[o * 576 + ci * 9 + kk];
    w2p[(size_t)n * 18432 + (((kk * 32 + (ci >> 1)) * 32 + o) << 1) + (ci & 1)] = v;
  }
}

// ---------------------------------------------------------------------------
// Modulated 3x3 conv (replicate pad) as implicit GEMM on V_WMMA_F32_16X16X4_F32.
// Each wave owns 16 out-channels x TWO 16-pixel tiles: one A fragment (b64
// global load) feeds two WMMAs. Patch is stored channel-fastest with stride
// CIN+4 so each B fragment is one aligned ds_load_b64 touching all 64 banks
// exactly once across the wave (no register repacking, no bank conflicts).
// Fused: noise inject (LDS channel reduction), bias, leaky ReLU, opt. toRGB.
// ---------------------------------------------------------------------------
template<int CIN, int COUT, int MT, int PT, bool FUSE_RGB>
__global__ __launch_bounds__(MT* PT * 16) void conv_mod_kernel(
    const float* __restrict__ xin, const float* __restrict__ wp,
    const float* __restrict__ noise,
    const float* __restrict__ gain_w, const float* __restrict__ gain_b,
    const float* __restrict__ bias,
    const float* __restrict__ rgb_w, const float* __restrict__ rgb_b,
    float* __restrict__ xout, float* __restrict__ rgbout)
{
  static_assert(PT % 2 == 0, "pixel tiles paired per wave");
  constexpr int PIX = PT * 16;          // pixels per workgroup (along W)
  constexpr int PW  = PIX + 2;          // padded patch width
  constexpr int CS  = CIN + 4;          // channel stride: 4 mod 64 banks
  constexpr int THREADS = MT * PT * 16; // (MT*PT/2) waves * 32 lanes
  __shared__ alignas(16) float patch[3 * PW * CS]; // [row][col][ci]
  __shared__ float gred[PIX];           // per-pixel noise-gain reduction
  __shared__ float rgbred[FUSE_RGB ? 3 * PIX : 1];

  const int t  = threadIdx.x;
  const int w0 = blockIdx.x * PIX, h = blockIdx.y, n = blockIdx.z;

  // Speculative prefetch of this sample's weight stream (global_prefetch_b8).
  __builtin_prefetch(wp + (size_t)n * 9 * (CIN / 2) * COUT * 2 + t, 0, 0);

  // Stage input patch rows h-1..h+1, cols w0-1..w0+PIX (replicate clamp).
  // Global reads coalesced along col; LDS writes transposed to [row][col][ci].
  for (int idx = t; idx < CIN * 3 * PW; idx += THREADS) {
    const int ci = idx / (3 * PW);
    const int rem = idx - ci * 3 * PW;
    const int r = rem / PW, col = rem - r * PW;
    const int hh = min(max(h - 1 + r, 0), H - 1);
    const int ww = min(max(w0 - 1 + col, 0), W - 1);
    patch[(r * PW + col) * CS + ci] =
        xin[(((size_t)n * CIN + ci) * H + hh) * W + ww];
  }
  for (int i = t; i < PIX; i += THREADS) gred[i] = 0.f;
  if (FUSE_RGB)
    for (int i = t; i < 3 * PIX; i += THREADS) rgbred[i] = 0.f;
  __syncthreads();

  const int wave = t >> 5, lane = t & 31;
  const int mtile = wave % MT;
  const int p0 = (wave / MT) * 2;       // first of the two pixel tiles
  const int halfsel = lane >> 4;        // K+2 half of the wave (A and B)
  const int lm = lane & 15;             // M index (A) / pixel index (B)

  v8f acc0 = {0.f, 0.f, 0.f, 0.f, 0.f, 0.f, 0.f, 0.f};
  v8f acc1 = {0.f, 0.f, 0.f, 0.f, 0.f, 0.f, 0.f, 0.f};

  for (int kh = 0; kh < 3; ++kh) {
    for (int kw = 0; kw < 3; ++kw) {
      const int kk = kh * 3 + kw;
      // A: w'[n][kk][ci/2][o][2]; lane reads (ci0, ci0+1) for o=mtile*16+lm
      const float* aptr =
          wp + (((((size_t)n * 9 + kk) * (CIN / 2)) + halfsel) * COUT +
                mtile * 16 + lm) * 2;
      // B: patch[kh][w0-1 + px + kw][ci0..ci0+1]; px = p*16+lm
      const float* bptr =
          &patch[(kh * PW + p0 * 16 + lm + kw) * CS + 2 * halfsel];
      #pragma unroll 4
      for (int k0 = 0; k0 < CIN; k0 += 4) {
        const v2f a = *(const v2f*)aptr;             // (ci0,ci0+1) b64 global
        aptr += 4 * COUT;                            // next 4 input channels
        const v2f b0 = *(const v2f*)bptr;            // ds_load_b64
        const v2f b1 = *(const v2f*)(bptr + 16 * CS); // ds_load_b64 (+imm off)
        bptr += 4;
        acc0 = __builtin_amdgcn_wmma_f32_16x16x4_f32(
            false, a, false, b0, (short)0, acc0, false, false);
        acc1 = __builtin_amdgcn_wmma_f32_16x16x4_f32(
            false, a, false, b1, (short)0, acc1, false, false);
      }
    }
  }

  // --- epilogue: per-pixel gain = sum_c gain_w[c]*conv[c] (LDS reduction) ---
  const int cbase = mtile * 16 + 8 * halfsel; // channels this lane holds
  float gw[8];
  #pragma unroll
  for (int r = 0; r < 8; ++r) gw[r] = gain_w[cbase + r];

  float part0 = 0.f, part1 = 0.f;
  #pragma unroll
  for (int r = 0; r < 8; ++r) { part0 += gw[r] * acc0[r]; part1 += gw[r] * acc1[r]; }
  atomicAdd(&gred[p0 * 16 + lm], part0);      // ds_add_f32
  atomicAdd(&gred[(p0 + 1) * 16 + lm], part1);
  __syncthreads();

  v8f accs[2] = {acc0, acc1};
  #pragma unroll
  for (int j = 0; j < 2; ++j) {
    const int pxg = (p0 + j) * 16 + lm;        // pixel within workgroup tile
    const float gain = gred[pxg] + gain_b[0];
    const float nz = noise[((size_t)n * H + h) * W + w0 + pxg];
    const float inj = nz * gain;
    const int wcol = w0 + pxg;

    float q0 = 0.f, q1 = 0.f, q2 = 0.f;
    #pragma unroll
    for (int r = 0; r < 8; ++r) {
      const int c = cbase + r;
      float v = accs[j][r] + inj + bias[c];
      v = (v >= 0.f) ? v : SLOPE * v;          // leaky ReLU
      xout[(((size_t)n * COUT + c) * H + h) * W + wcol] = v;
      if (FUSE_RGB) {
        q0 += rgb_w[c] * v;
        q1 += rgb_w[COUT + c] * v;
        q2 += rgb_w[2 * COUT + c] * v;
      }
    }
    if (FUSE_RGB) {
      atomicAdd(&rgbred[pxg], q0);
      atomicAdd(&rgbred[PIX + pxg], q1);
      atomicAdd(&rgbred[2 * PIX + pxg], q2);
    }
  }

  if (FUSE_RGB) {
    __syncthreads();
    for (int i = t; i < 3 * PIX; i += THREADS) {
      const int o = i / PIX, px = i - o * PIX;
      rgbout[(((size_t)n * 3 + o) * H + h) * W + w0 + px] = rgbred[i] + rgb_b[o];
    }
  }
}

// ---------------------------------------------------------------------------
extern "C" void kernel_launch(void* const* d_in, const int* in_sizes, int n_in,
                              void* d_out, int out_size, void* d_ws, size_t ws_size,
                              hipStream_t stream) {
  const float* x    = (const float*)d_in[0];
  const float* y    = (const float*)d_in[1];
  const float* nz1  = (const float*)d_in[2];
  const float* nz2  = (const float*)d_in[3];
  const float* a1w  = (const float*)d_in[4];
  const float* a1b  = (const float*)d_in[5];
  const float* c1w  = (const float*)d_in[6];
  const float* b1   = (const float*)d_in[7];
  const float* g1w  = (const float*)d_in[8];
  const float* g1b  = (const float*)d_in[9];
  const float* a2w  = (const float*)d_in[10];
  const float* a2b  = (const float*)d_in[11];
  const float* c2w  = (const float*)d_in[12];
  const float* b2   = (const float*)d_in[13];
  const float* g2w  = (const float*)d_in[14];
  const float* g2b  = (const float*)d_in[15];
  const float* rgbw = (const float*)d_in[16];
  const float* rgbb = (const float*)d_in[17];

  float* ws   = (float*)d_ws;
  float* xmid = ws;                                  // 4*64*512*512 floats
  float* w1p  = ws + (size_t)NB * 64 * H * W;        // 147456 floats
  float* w2p  = w1p + 147456;                        // 73728 floats

  float* xo   = (float*)d_out;                       // x: 4*32*512*512
  float* rgbo = xo + (size_t)NB * 32 * H * W;        // rgb: 4*3*512*512

  prep_kernel<<<dim3(NB), 256, 0, stream>>>(y, a1w, a1b, c1w, a2w, a2b, c2w,
                                            w1p, w2p);

  // conv1: 64->64, MT=4 x PT=4 (64 px/WG), 8 waves, 2 px-tiles per wave
  conv_mod_kernel<64, 64, 4, 4, false>
      <<<dim3(W / 64, H, NB), 256, 0, stream>>>(
          x, w1p, nz1, g1w, g1b, b1, nullptr, nullptr, xmid, nullptr);

  // conv2: 64->32, MT=2 x PT=4 (64 px/WG), 4 waves, 2 px-tiles/wave, fused toRGB
  conv_mod_kernel<64, 32, 2, 4, true>
      <<<dim3(W / 64, H, NB), 128, 0, stream>>>(
          xmid, w2p, nz2, g2w, g2b, b2, rgbw, rgbb, xo, rgbo);
}